// QuantumAttentionLayer_24223615550297
// MI455X (gfx1250) — compile-verified
//
#include <hip/hip_runtime.h>
#include <cmath>

// ---------------------------------------------------------------------------
// QuantumAttentionLayer for MI455X (gfx1250, wave32)
//   Stage A: proj GEMM (2048x1024)x(1024x20) + tanh     -> V_WMMA_F32_16X16X4_F32
//   Stage B: 2048 independent 10-qubit statevector sims -> LDS-resident, 512 thr/row
//   Stage C: (16384x10)x(10x1024) + bias + residual     -> bandwidth-bound VALU
// ---------------------------------------------------------------------------

#define NQ        10
#define NL        3
#define BB        8
#define SS        2048
#define DD        1024
#define NSUB      256
#define MROWS     (BB * NSUB)     // 2048 encoded rows
#define ENC_COLS  (2 * NQ)        // 20

typedef float v2f __attribute__((ext_vector_type(2)));
typedef float v8f __attribute__((ext_vector_type(8)));

// ---------------------------------------------------------------------------
// Stage A: enc[m][j] = tanh( sum_k x[b, 8*ss, k] * Wp[j][k] + bp[j] )
// One wave (32 threads) per 16x16 output tile. grid = (128 M-tiles, 2 N-tiles).
// WMMA f32 16x16x4 fragment layout (ISA 7.12.2):
//   A (16x4): lane<16 row M=lane holds K={k,k+1}; lane>=16 holds K={k+2,k+3}
//   B (4x16): lane<16 col N=lane holds K={k,k+1}; lane>=16 holds K={k+2,k+3}
//   C (16x16): VGPR v, lanes<16 -> M=v, lanes>=16 -> M=v+8; N = lane&15
// Out-of-range N columns (20..31) are handled branch-free: the lane loads a
// safe row (Wp row 0) and multiplies by a 0/1 mask, keeping EXEC uniform so
// the inner loop is pure global_load_b64 -> v_wmma with no saveexec traffic.
// ---------------------------------------------------------------------------
__global__ __launch_bounds__(32)
void proj_kernel(const float* __restrict__ x, const float* __restrict__ Wp,
                 const float* __restrict__ bp, float* __restrict__ enc) {
    const int lane = threadIdx.x;
    const int lo = lane & 15;
    const int hi = lane >> 4;

    // A-fragment row: subsampled x row (every 8th seq position)
    const int mrow = blockIdx.x * 16 + lo;
    const float* xrow = x + ((size_t)(mrow >> 8) * SS + (size_t)(mrow & 255) * 8) * DD;

    // B-fragment column: row of Wp (enc = x @ Wp^T)
    const int ncol = blockIdx.y * 16 + lo;
    const bool nvalid = (ncol < ENC_COLS);
    const float bmask = nvalid ? 1.0f : 0.0f;
    const float* wrow = Wp + (size_t)(nvalid ? ncol : 0) * DD;  // safe addr always

    float cacc[8];

#if __has_builtin(__builtin_amdgcn_wmma_f32_16x16x4_f32)
    v8f c = {0.f, 0.f, 0.f, 0.f, 0.f, 0.f, 0.f, 0.f};
    for (int k = 0; k < DD; k += 4) {
        const float2 av = *(const float2*)(xrow + k + 2 * hi);
        const float2 bv = *(const float2*)(wrow + k + 2 * hi);
        v2f a, b;
        a.x = av.x;         a.y = av.y;
        b.x = bv.x * bmask; b.y = bv.y * bmask;
        // (neg_a, A, neg_b, B, c_mod, C, reuse_a, reuse_b)
        c = __builtin_amdgcn_wmma_f32_16x16x4_f32(false, a, false, b,
                                                  (short)0, c, false, false);
    }
#pragma unroll
    for (int v = 0; v < 8; ++v) cacc[v] = c[v];
#else
    // VALU fallback (kept correct; histogram will reveal if this path was taken)
#pragma unroll
    for (int v = 0; v < 8; ++v) cacc[v] = 0.f;
    for (int k = 0; k < DD; ++k) {
        const float wv = wrow[k] * bmask;
#pragma unroll
        for (int v = 0; v < 8; ++v) {
            const int row = blockIdx.x * 16 + v + 8 * hi;
            const float* xr =
                x + ((size_t)(row >> 8) * SS + (size_t)(row & 255) * 8) * DD;
            cacc[v] += xr[k] * wv;
        }
    }
#endif

#pragma unroll
    for (int v = 0; v < 8; ++v) {
        const int row = blockIdx.x * 16 + v + 8 * hi;
        const int col = blockIdx.y * 16 + lo;
        if (col < ENC_COLS)
            enc[row * ENC_COLS + col] = tanhf(cacc[v] + bp[col]);
    }
}

// ---------------------------------------------------------------------------
// Stage B: 10-qubit statevector simulation, one block per enc row.
// State: 1024 complex amplitudes in LDS. Qubit i <-> bit (9-i) of the index.
// ---------------------------------------------------------------------------
__device__ __forceinline__
void apply_gate(float2* psi, int t, int bit,
                float2 u00, float2 u01, float2 u10, float2 u11) {
    const int mask = (1 << bit) - 1;
    const int k0 = ((t & ~mask) << 1) | (t & mask);
    const int k1 = k0 | (1 << bit);
    const float2 a0 = psi[k0];
    const float2 a1 = psi[k1];
    float2 r0, r1;
    r0.x = u00.x * a0.x - u00.y * a0.y + u01.x * a1.x - u01.y * a1.y;
    r0.y = u00.x * a0.y + u00.y * a0.x + u01.x * a1.y + u01.y * a1.x;
    r1.x = u10.x * a0.x - u10.y * a0.y + u11.x * a1.x - u11.y * a1.y;
    r1.y = u10.x * a0.y + u10.y * a0.x + u11.x * a1.y + u11.y * a1.x;
    psi[k0] = r0;
    psi[k1] = r1;
}

__global__ __launch_bounds__(512)
void vqc_kernel(const float* __restrict__ enc, const float* __restrict__ qw,
                const float* __restrict__ qscale, float* __restrict__ xq) {
    __shared__ float2 psi[1 << NQ];          // 8 KB statevector
    __shared__ float  red[16 * NQ];          // per-wave partial expvals

    const int t = threadIdx.x;
    const int row = blockIdx.x;

    psi[t]       = make_float2(0.f, 0.f);
    psi[t + 512] = make_float2(0.f, 0.f);
    if (t == 0) psi[0] = make_float2(1.f, 0.f);
    __syncthreads();

    const float* erow = enc + row * ENC_COLS;
    const float PI = 3.14159265358979323846f;

    for (int layer = 0; layer < NL; ++layer) {
        // ---- encoding gates: U = RZ(e1*pi) * RY(e0*pi) ----
        for (int i = 0; i < NQ; ++i) {
            const float th = erow[2 * i]     * PI * 0.5f;  // theta/2
            const float ph = erow[2 * i + 1] * PI * 0.5f;  // phi/2
            float cth, sth, cph, sph;
            sincosf(th, &sth, &cth);
            sincosf(ph, &sph, &cph);
            const float2 u00 = make_float2( cth * cph, -cth * sph);
            const float2 u01 = make_float2(-sth * cph,  sth * sph);
            const float2 u10 = make_float2( sth * cph,  sth * sph);
            const float2 u11 = make_float2( cth * cph,  cth * sph);
            apply_gate(psi, t, 9 - i, u00, u01, u10, u11);
            __syncthreads();
        }
        // ---- trained gates: U = RZ(w2) * RY(w1) * RZ(w0) ----
        for (int i = 0; i < NQ; ++i) {
            const float* w = qw + (layer * NQ + i) * 3;
            float c1, s1;
            sincosf(w[1] * 0.5f, &s1, &c1);
            const float Ah = (w[0] + w[2]) * 0.5f;
            const float Bh = (w[0] - w[2]) * 0.5f;
            float cA, sA, cB, sB;
            sincosf(Ah, &sA, &cA);
            sincosf(Bh, &sB, &cB);
            const float2 u00 = make_float2( c1 * cA, -c1 * sA);  //  c*e^{-i(w0+w2)/2}
            const float2 u01 = make_float2(-s1 * cB, -s1 * sB);  // -s*e^{+i(w0-w2)/2}
            const float2 u10 = make_float2( s1 * cB, -s1 * sB);  //  s*e^{-i(w0-w2)/2}
            const float2 u11 = make_float2( c1 * cA,  c1 * sA);  //  c*e^{+i(w0+w2)/2}
            apply_gate(psi, t, 9 - i, u00, u01, u10, u11);
            __syncthreads();
        }
        // ---- CNOT ring: control i, target (i+1)%10 ----
        for (int i = 0; i < NQ; ++i) {
            if (t < 256) {
                const int bc = 9 - i;
                const int bt = 9 - ((i + 1) % NQ);
                const int bl = bc < bt ? bc : bt;
                const int bh = bc ^ bt ^ bl;
                const int ml = (1 << bl) - 1;
                const int tmp = ((t & ~ml) << 1) | (t & ml);
                const int mh = (1 << bh) - 1;
                const int base = ((tmp & ~mh) << 1) | (tmp & mh);
                const int k10 = base | (1 << bc);
                const int k11 = k10 | (1 << bt);
                const float2 tv = psi[k10];
                psi[k10] = psi[k11];
                psi[k11] = tv;
            }
            __syncthreads();
        }
    }

    // ---- PauliZ expvals: out[q] = sum_k |psi_k|^2 * (1 - 2*bit(9-q)(k)) ----
    float acc[NQ];
#pragma unroll
    for (int q = 0; q < NQ; ++q) acc[q] = 0.f;
#pragma unroll
    for (int half = 0; half < 2; ++half) {
        const int k = t + half * 512;
        const float2 a = psi[k];
        const float p = a.x * a.x + a.y * a.y;
#pragma unroll
        for (int q = 0; q < NQ; ++q)
            acc[q] += ((k >> (9 - q)) & 1) ? -p : p;
    }
    // wave32 butterfly reduction
#pragma unroll
    for (int q = 0; q < NQ; ++q) {
        for (int off = 16; off > 0; off >>= 1)
            acc[q] += __shfl_xor(acc[q], off, 32);
    }
    const int wv = t >> 5;
    if ((t & 31) == 0) {
#pragma unroll
        for (int q = 0; q < NQ; ++q) red[wv * NQ + q] = acc[q];
    }
    __syncthreads();
    if (t < NQ) {
        float s = 0.f;
#pragma unroll
        for (int w = 0; w < 16; ++w) s += red[w * NQ + t];
        xq[row * NQ + t] = s * qscale[0];
    }
}

// ---------------------------------------------------------------------------
// Stage C: out[b,s,:] = lerp(xq) @ Wo^T + bo + x[b,s,:]
// One block per (b,s) row; 256 threads x float4. Pure bandwidth (~128 MB).
// ---------------------------------------------------------------------------
__global__ __launch_bounds__(256)
void out_kernel(const float* __restrict__ x, const float* __restrict__ xq,
                const float* __restrict__ Wo, const float* __restrict__ bo,
                float* __restrict__ out) {
    const int bs = blockIdx.x;
    const int b = bs >> 11;                 // S = 2048
    const int s = bs & (SS - 1);

    float src = (s + 0.5f) * (1.0f / 8.0f) - 0.5f;
    src = fminf(fmaxf(src, 0.f), (float)(NSUB - 1));
    const int i0 = (int)src;                // src >= 0 -> trunc == floor
    const float lam = src - (float)i0;
    const int i1 = min(i0 + 1, NSUB - 1);

    const float* q0 = xq + (size_t)(b * NSUB + i0) * NQ;
    const float* q1 = xq + (size_t)(b * NSUB + i1) * NQ;
    float qv[NQ];
#pragma unroll
    for (int j = 0; j < NQ; ++j)
        qv[j] = (1.f - lam) * q0[j] + lam * q1[j];

    const size_t rowoff = (size_t)bs * DD;
    const int d0 = threadIdx.x * 4;
    const float4 xv = *(const float4*)(x + rowoff + d0);

    float r[4];
#pragma unroll
    for (int dd = 0; dd < 4; ++dd) {
        const int d = d0 + dd;
        float a = bo[d];
        const float* wrow = Wo + (size_t)d * NQ;
#pragma unroll
        for (int j = 0; j < NQ; ++j) a += qv[j] * wrow[j];
        r[dd] = a;
    }
    float4 res;
    res.x = r[0] + xv.x;
    res.y = r[1] + xv.y;
    res.z = r[2] + xv.z;
    res.w = r[3] + xv.w;
    *(float4*)(out + rowoff + d0) = res;
}

// ---------------------------------------------------------------------------
extern "C" void kernel_launch(void* const* d_in, const int* in_sizes, int n_in,
                              void* d_out, int out_size, void* d_ws, size_t ws_size,
                              hipStream_t stream) {
    const float* x  = (const float*)d_in[0];   // (8, 2048, 1024)
    const float* Wp = (const float*)d_in[1];   // (20, 1024)
    const float* bp = (const float*)d_in[2];   // (20,)
    const float* qw = (const float*)d_in[3];   // (3, 10, 3)
    const float* qs = (const float*)d_in[4];   // (1,)
    const float* Wo = (const float*)d_in[5];   // (1024, 10)
    const float* bo = (const float*)d_in[6];   // (1024,)
    float* out = (float*)d_out;                // (8, 2048, 1024)

    float* enc = (float*)d_ws;                 // 2048 * 20 floats
    float* xq  = enc + MROWS * ENC_COLS;       // 2048 * 10 floats

    proj_kernel<<<dim3(MROWS / 16, 2), 32, 0, stream>>>(x, Wp, bp, enc);
    vqc_kernel<<<MROWS, 512, 0, stream>>>(enc, qw, qs, xq);
    out_kernel<<<BB * SS, 256, 0, stream>>>(x, xq, Wo, bo, out);
}